// graph_gru_gcn_80333068304418
// MI455X (gfx1250) — compile-verified
//
#include <hip/hip_runtime.h>
#include <math.h>

#define DFEAT 128
#define WPK_FLOATS 16384   // packed weight matrix: 32 kblk * 4 tpair * 32 lane * 4 = 64KB
#define LDS_PAD_F4 132     // pad so the pipelined B overfetch (kb==32) stays in LDS

typedef float v2f __attribute__((ext_vector_type(2)));
typedef float v8f __attribute__((ext_vector_type(8)));

// ---------------------------------------------------------------------------
// GCN degree normalization: deg[i] = 1 (self loop) + sum_{e: col[e]==i} w[e]
// then dinv[i] = rsqrt(deg[i])
// ---------------------------------------------------------------------------
__global__ void k_deg_init(float* __restrict__ deg, int N) {
  int stride = gridDim.x * blockDim.x;
  for (int i = blockIdx.x * blockDim.x + threadIdx.x; i < N; i += stride)
    deg[i] = 1.0f;  // self-loop weight
}

__global__ void k_deg_acc(const int* __restrict__ col, const float* __restrict__ w,
                          float* __restrict__ deg, int E) {
  int stride = gridDim.x * blockDim.x;
  for (int i = blockIdx.x * blockDim.x + threadIdx.x; i < E; i += stride)
    atomicAdd(&deg[col[i]], w[i]);
}

__global__ void k_dinv(float* __restrict__ deg, int N) {
  int stride = gridDim.x * blockDim.x;
  for (int i = blockIdx.x * blockDim.x + threadIdx.x; i < N; i += stride) {
    float d = deg[i];
    deg[i] = (d > 0.0f) ? rsqrtf(d) : 0.0f;
  }
}

// ---------------------------------------------------------------------------
// Pre-pack a 128x128 weight matrix into the WMMA B-fragment layout so the
// GEMM can fetch each lane's fragments for TWO column tiles with one b128.
// float4 index: idx = (kb*4 + tp)*32 + lane
//   k0 = kb*4 + (lane>=16 ? 2 : 0)      (B 4x16 layout: VGPR v = row K=v / K=v+2)
//   n0 = 32*tp + (lane&15), n1 = n0+16  (tiles 2tp and 2tp+1)
//   v  = { W[k0][n0], W[k0+1][n0], W[k0][n1], W[k0+1][n1] }
// ---------------------------------------------------------------------------
__global__ void k_pack_w(const float* __restrict__ W, float* __restrict__ Wp) {
  int idx = blockIdx.x * blockDim.x + threadIdx.x;  // 0..4095
  if (idx >= WPK_FLOATS / 4) return;
  int lane = idx & 31;
  int tp   = (idx >> 5) & 3;
  int kb   = idx >> 7;
  int k0   = kb * 4 + ((lane >> 4) << 1);
  int n0   = 32 * tp + (lane & 15);
  float4 v;
  v.x = W[(k0 + 0) * DFEAT + n0];
  v.y = W[(k0 + 1) * DFEAT + n0];
  v.z = W[(k0 + 0) * DFEAT + n0 + 16];
  v.w = W[(k0 + 1) * DFEAT + n0 + 16];
  ((float4*)Wp)[idx] = v;
}

// ---------------------------------------------------------------------------
// Fused dual GEMM: Out[N,128] = Xa @ Wa + Xb @ Wb  (fp32, V_WMMA_F32_16X16X4_F32)
// Packed weights staged in LDS; one wave per 16-row tile; 8 accumulator tiles
// cover all 128 output columns. Explicit 2-stage software pipeline: fragment
// buffers b0/b1 are simultaneously live, so loads for k-step+1 overlap the 8
// WMMAs of the current k-step instead of stalling on s_wait_dscnt each pair.
// ---------------------------------------------------------------------------
__device__ __forceinline__ void gd_frag_load(float4* __restrict__ dst,
                                             const float4* __restrict__ wp,
                                             int kb, int lane) {
#pragma unroll
  for (int tp = 0; tp < 4; ++tp) dst[tp] = wp[(kb * 4 + tp) * 32 + lane];
}

__device__ __forceinline__ void gd_compute(v8f* __restrict__ acc, v2f a,
                                           const float4* __restrict__ bv) {
#pragma unroll
  for (int tp = 0; tp < 4; ++tp) {
    v2f b0; b0.x = bv[tp].x; b0.y = bv[tp].y;   // tile 2tp
    v2f b1; b1.x = bv[tp].z; b1.y = bv[tp].w;   // tile 2tp+1
    acc[2 * tp] = __builtin_amdgcn_wmma_f32_16x16x4_f32(
        false, a, false, b0, (short)0, acc[2 * tp], false, false);
    acc[2 * tp + 1] = __builtin_amdgcn_wmma_f32_16x16x4_f32(
        false, a, false, b1, (short)0, acc[2 * tp + 1], false, false);
  }
}

__global__ __launch_bounds__(128)
void k_gemm_dual(const float* __restrict__ Xa, const float* __restrict__ Wpa,
                 const float* __restrict__ Xb, const float* __restrict__ Wpb,
                 float* __restrict__ Out, int N) {
  extern __shared__ float4 ldsW[];  // 2*4096 float4 + pad

  // cooperative stage of both packed weight matrices into LDS
  for (int i = threadIdx.x; i < WPK_FLOATS / 4; i += 128) {
    ldsW[i]                  = ((const float4*)Wpa)[i];
    ldsW[i + WPK_FLOATS / 4] = ((const float4*)Wpb)[i];
  }
  __syncthreads();

  const int lane = threadIdx.x & 31;
  const int wave = threadIdx.x >> 5;
  const int m0 = (blockIdx.x * 4 + wave) * 16;
  if (m0 >= N) return;  // wave-uniform exit (EXEC all-ones for WMMA)

  v8f acc[8] = {};
  const int rsel  = lane & 15;
  const int khalf = (lane >> 4) << 1;

  for (int s = 0; s < 2; ++s) {
    const float* xrow = (s ? Xb : Xa) + (size_t)(m0 + rsel) * DFEAT;
    const float4* wp  = ldsW + s * (WPK_FLOATS / 4);

    float4 fb0[4], fb1[4];
    v2f a0, a1;
    gd_frag_load(fb0, wp, 0, lane);
    a0.x = xrow[khalf];
    a0.y = xrow[khalf + 1];

#pragma unroll 1
    for (int kb = 0; kb < 32; kb += 2) {
      // stage k-step kb+1 while computing kb
      a1.x = xrow[(kb + 1) * 4 + khalf];
      a1.y = xrow[(kb + 1) * 4 + khalf + 1];
      gd_frag_load(fb1, wp, kb + 1, lane);
      gd_compute(acc, a0, fb0);

      // stage k-step kb+2 while computing kb+1 (tail overfetch stays
      // in padded LDS; A index clamped to remain in-bounds globally)
      int kn = (kb + 2 < 32) ? (kb + 2) : 31;
      a0.x = xrow[kn * 4 + khalf];
      a0.y = xrow[kn * 4 + khalf + 1];
      gd_frag_load(fb0, wp, kb + 2, lane);
      gd_compute(acc, a1, fb1);
    }
  }

  // C/D layout: VGPR v, lanes 0-15 -> row m0+v; lanes 16-31 -> row m0+8+v
  const int srow = m0 + (lane >> 4) * 8;
#pragma unroll
  for (int v = 0; v < 8; ++v) {
    float* orow = Out + (size_t)(srow + v) * DFEAT + rsel;
#pragma unroll
    for (int t = 0; t < 8; ++t) orow[t * 16] = acc[t][v];
  }
}

// ---------------------------------------------------------------------------
// Dual aggregation init: AggU = dinv^2*U + bzx+bzh ; AggV = dinv^2*V + brx+brh
// ---------------------------------------------------------------------------
__global__ void k_agg_init2(float* __restrict__ AggU, const float* __restrict__ U,
                            float* __restrict__ AggV, const float* __restrict__ V,
                            const float* __restrict__ dinv,
                            const float* __restrict__ bu1, const float* __restrict__ bu2,
                            const float* __restrict__ bv1, const float* __restrict__ bv2,
                            int N) {
  int total = N * DFEAT;
  int stride = gridDim.x * blockDim.x;
  for (int idx = blockIdx.x * blockDim.x + threadIdx.x; idx < total; idx += stride) {
    int i = idx >> 7;
    int d = idx & (DFEAT - 1);
    float di = dinv[i];
    float di2 = di * di;
    AggU[idx] = di2 * U[idx] + bu1[d] + bu2[d];
    AggV[idx] = di2 * V[idx] + bv1[d] + bv2[d];
  }
}

__global__ void k_agg_init(float* __restrict__ Agg, const float* __restrict__ U,
                           const float* __restrict__ dinv,
                           const float* __restrict__ b1, const float* __restrict__ b2,
                           int N) {
  int total = N * DFEAT;
  int stride = gridDim.x * blockDim.x;
  for (int idx = blockIdx.x * blockDim.x + threadIdx.x; idx < total; idx += stride) {
    int i = idx >> 7;
    int d = idx & (DFEAT - 1);
    float di = dinv[i];
    Agg[idx] = di * di * U[idx] + b1[d] + b2[d];
  }
}

// ---------------------------------------------------------------------------
// Dual edge scatter (z and r gates share the norm and index reads):
//   AggU[c,:] += nrm*U[r,:] ; AggV[c,:] += nrm*V[r,:]
// One wave per edge; 32 lanes x float4 = 128 features.
// ---------------------------------------------------------------------------
__global__ __launch_bounds__(256)
void k_agg_edges2(const int* __restrict__ rows, const int* __restrict__ cols,
                  const float* __restrict__ w, const float* __restrict__ dinv,
                  const float* __restrict__ U, float* __restrict__ AggU,
                  const float* __restrict__ V, float* __restrict__ AggV, int E) {
  const int lane = threadIdx.x & 31;
  const int wid = (blockIdx.x * blockDim.x + threadIdx.x) >> 5;
  const int nwaves = (gridDim.x * blockDim.x) >> 5;
  for (int e = wid; e < E; e += nwaves) {
    const int r = rows[e];
    const int c = cols[e];
    const float nrm = dinv[r] * w[e] * dinv[c];
    const float4 u = ((const float4*)(U + (size_t)r * DFEAT))[lane];
    const float4 v = ((const float4*)(V + (size_t)r * DFEAT))[lane];
    float* du = AggU + (size_t)c * DFEAT + lane * 4;
    float* dv = AggV + (size_t)c * DFEAT + lane * 4;
    atomicAdd(du + 0, nrm * u.x);
    atomicAdd(du + 1, nrm * u.y);
    atomicAdd(du + 2, nrm * u.z);
    atomicAdd(du + 3, nrm * u.w);
    atomicAdd(dv + 0, nrm * v.x);
    atomicAdd(dv + 1, nrm * v.y);
    atomicAdd(dv + 2, nrm * v.z);
    atomicAdd(dv + 3, nrm * v.w);
  }
}

__global__ __launch_bounds__(256)
void k_agg_edges(const int* __restrict__ rows, const int* __restrict__ cols,
                 const float* __restrict__ w, const float* __restrict__ dinv,
                 const float* __restrict__ U, float* __restrict__ Agg, int E) {
  const int lane = threadIdx.x & 31;
  const int wid = (blockIdx.x * blockDim.x + threadIdx.x) >> 5;
  const int nwaves = (gridDim.x * blockDim.x) >> 5;
  for (int e = wid; e < E; e += nwaves) {
    const int r = rows[e];
    const int c = cols[e];
    const float nrm = dinv[r] * w[e] * dinv[c];
    const float4 u = ((const float4*)(U + (size_t)r * DFEAT))[lane];
    float* dst = Agg + (size_t)c * DFEAT + lane * 4;
    atomicAdd(dst + 0, nrm * u.x);
    atomicAdd(dst + 1, nrm * u.y);
    atomicAdd(dst + 2, nrm * u.z);
    atomicAdd(dst + 3, nrm * u.w);
  }
}

// ---------------------------------------------------------------------------
// Elementwise: z = sigmoid(z_pre) [in place], rh = sigmoid(r_pre) * h [in place]
// ---------------------------------------------------------------------------
__global__ void k_ew_zr(float* __restrict__ zio, float* __restrict__ rio,
                        const float* __restrict__ h, int total) {
  int stride = gridDim.x * blockDim.x;
  for (int idx = blockIdx.x * blockDim.x + threadIdx.x; idx < total; idx += stride) {
    float z = 1.0f / (1.0f + __expf(-zio[idx]));
    float r = 1.0f / (1.0f + __expf(-rio[idx]));
    zio[idx] = z;
    rio[idx] = r * h[idx];
  }
}

// ---------------------------------------------------------------------------
// Elementwise output: ho = z*h + (1-z)*tanh(t_pre)
// ---------------------------------------------------------------------------
__global__ void k_ew_out(float* __restrict__ ho, const float* __restrict__ z,
                         const float* __restrict__ h, const float* __restrict__ tpre,
                         int total) {
  int stride = gridDim.x * blockDim.x;
  for (int idx = blockIdx.x * blockDim.x + threadIdx.x; idx < total; idx += stride) {
    float zz = z[idx];
    float ht = tanhf(tpre[idx]);
    ho[idx] = zz * h[idx] + (1.0f - zz) * ht;
  }
}

static inline int imin(int a, int b) { return a < b ? a : b; }

extern "C" void kernel_launch(void* const* d_in, const int* in_sizes, int n_in,
                              void* d_out, int out_size, void* d_ws, size_t ws_size,
                              hipStream_t stream) {
  const float* x   = (const float*)d_in[0];
  const int*   ei  = (const int*)d_in[1];
  const float* ewt = (const float*)d_in[2];
  const float* h   = (const float*)d_in[3];
  const float* Wxz = (const float*)d_in[4];  const float* bxz = (const float*)d_in[5];
  const float* Whz = (const float*)d_in[6];  const float* bhz = (const float*)d_in[7];
  const float* Wxr = (const float*)d_in[8];  const float* bxr = (const float*)d_in[9];
  const float* Whr = (const float*)d_in[10]; const float* bhr = (const float*)d_in[11];
  const float* Wxh = (const float*)d_in[12]; const float* bxh = (const float*)d_in[13];
  const float* Whh = (const float*)d_in[14]; const float* bhh = (const float*)d_in[15];
  float* out = (float*)d_out;

  const int D = DFEAT;
  const int N = in_sizes[0] / D;               // 50000
  const int E = in_sizes[2];                   // 1.6M
  const int L = in_sizes[3] / (N * D);         // 2
  const size_t ND = (size_t)N * D;
  const size_t DD = (size_t)D * D;

  // Workspace: dinv[N] (padded) + 4 x [N,D] buffers + 6L packed weight mats
  float* ws   = (float*)d_ws;
  float* dinv = ws;
  float* buf0 = ws + (((size_t)N + 255) & ~(size_t)255);
  float* buf1 = buf0 + ND;
  float* buf2 = buf1 + ND;
  float* buf3 = buf2 + ND;
  float* wpk  = buf3 + ND;   // L * 6 * WPK_FLOATS

  const int* rows = ei;       // edge_index[0] = source
  const int* cols = ei + E;   // edge_index[1] = target

  const int gN    = imin((N + 255) / 256, 4096);
  const int gE    = imin((E + 255) / 256, 4096);
  const int gND   = imin((int)((ND + 255) / 256), 4096);
  const int gEdge = 4096;
  const int gGemm = (N + 63) / 64;
  const int gPack = (WPK_FLOATS / 4 + 255) / 256;
  const size_t ldsBytes = (2 * (size_t)(WPK_FLOATS / 4) + LDS_PAD_F4) * sizeof(float4);

  // Pack all weights into WMMA fragment layout (slots: xz,hz,xr,hr,xh,hh)
  const float* Wsrc[6] = {Wxz, Whz, Wxr, Whr, Wxh, Whh};
  for (int i = 0; i < L; ++i)
    for (int s = 0; s < 6; ++s)
      k_pack_w<<<gPack, 256, 0, stream>>>(Wsrc[s] + i * DD,
                                          wpk + ((size_t)i * 6 + s) * WPK_FLOATS);

  // Shared GCN normalization (one pass, reused by all 12 convs)
  k_deg_init<<<gN, 256, 0, stream>>>(dinv, N);
  k_deg_acc <<<gE, 256, 0, stream>>>(cols, ewt, dinv, E);
  k_dinv    <<<gN, 256, 0, stream>>>(dinv, N);

  for (int i = 0; i < L; ++i) {
    const float* xin = (i == 0) ? x : out + (size_t)(i - 1) * ND;
    const float* hi  = h + (size_t)i * ND;
    const float* wp  = wpk + (size_t)i * 6 * WPK_FLOATS;

    // z pre-GEMM: buf0 = xin@Wxz + h@Whz ; r pre-GEMM: buf1 = xin@Wxr + h@Whr
    k_gemm_dual<<<gGemm, 128, ldsBytes, stream>>>(xin, wp + 0 * WPK_FLOATS,
                                                  hi,  wp + 1 * WPK_FLOATS, buf0, N);
    k_gemm_dual<<<gGemm, 128, ldsBytes, stream>>>(xin, wp + 2 * WPK_FLOATS,
                                                  hi,  wp + 3 * WPK_FLOATS, buf1, N);

    // fused z+r aggregation (self-loop + bias init, then one edge pass)
    k_agg_init2 <<<gND, 256, 0, stream>>>(buf2, buf0, buf3, buf1, dinv,
                                          bxz + i * D, bhz + i * D,
                                          bxr + i * D, bhr + i * D, N);
    k_agg_edges2<<<gEdge, 256, 0, stream>>>(rows, cols, ewt, dinv,
                                            buf0, buf2, buf1, buf3, E);

    // z = sigmoid(buf2) in place; buf3 = sigmoid(buf3) * h  (= r*h)
    k_ew_zr<<<gND, 256, 0, stream>>>(buf2, buf3, hi, (int)ND);

    // candidate pre-GEMM: buf0 = xin@Wxh + (r*h)@Whh ; aggregate into buf1
    k_gemm_dual<<<gGemm, 128, ldsBytes, stream>>>(xin, wp + 4 * WPK_FLOATS,
                                                  buf3, wp + 5 * WPK_FLOATS, buf0, N);
    k_agg_init <<<gND, 256, 0, stream>>>(buf1, buf0, dinv, bxh + i * D, bhh + i * D, N);
    k_agg_edges<<<gEdge, 256, 0, stream>>>(rows, cols, ewt, dinv, buf0, buf1, E);

    // h_out = z*h + (1-z)*tanh(buf1)
    k_ew_out<<<gND, 256, 0, stream>>>(out + (size_t)i * ND, buf2, hi, buf1, (int)ND);
  }
}